// DifferentiableHodgeProxy_60060822667821
// MI455X (gfx1250) — compile-verified
//
#include <hip/hip_runtime.h>
#include <math.h>

// DifferentiableHodgeProxy for MI455X (gfx1250, wave32, WMMA + TDM).
#define KPTS 16
#define NE   120           // C(16,2) edges
#define NSC  3
#define TAU_C 0.0001f
#define HID  512
#define NPTS 8192
#define FDIM_PAD 32        // 28 real features, zero-padded to 32 for WMMA K

typedef float v2f __attribute__((ext_vector_type(2)));
typedef float v8f __attribute__((ext_vector_type(8)));
typedef unsigned int u32x4 __attribute__((ext_vector_type(4)));
typedef int i32x4 __attribute__((ext_vector_type(4)));
typedef int i32x8 __attribute__((ext_vector_type(8)));

__device__ __forceinline__ int e2i(int i, int j) {   // edge index, i<j, K=16
  return i * (31 - i) / 2 + (j - i - 1);
}

// ---------------------------------------------------------------------------
// TDM helper: 2D tile load Global -> LDS (tensor dims == tile dims, all
// in-bounds; dim0 contiguous elements, dim1 rows of stride0 f32 elements).
// D# packing per CDNA5 ISA 8.3/8.4: group0 = {flags, lds_addr, gaddr_lo,
// gaddr_hi|type=2}; group1 carries data_size=4B, tensor/tile dims, stride.
// This toolchain exposes the 6-arg builtin (extra i32x8 group before cpol).
// ---------------------------------------------------------------------------
__device__ __forceinline__ void tdm_load_2d_f32(void* lds_dst, const void* gsrc,
                                                unsigned tile0, unsigned tile1,
                                                unsigned stride0_elems) {
  unsigned lds_addr = (unsigned)(unsigned long long)(uintptr_t)lds_dst; // LDS offset = low 32b of flat
  unsigned long long ga = (unsigned long long)(uintptr_t)gsrc;
  u32x4 g0;
  g0[0] = 1u;                                               // count=1, user descriptor
  g0[1] = lds_addr;                                         // bits 63:32
  g0[2] = (unsigned)ga;                                     // bits 95:64
  g0[3] = (unsigned)((ga >> 32) & 0x1FFFFFFu) | (2u << 30); // addr[56:32] | type=2
  i32x8 g1;
  g1[0] = (int)(2u << 16);                                  // data_size=2 (4 bytes)
  g1[1] = (int)((tile0 & 0xFFFFu) << 16);                   // tensor_dim0[15:0] @ bit48
  g1[2] = (int)(((tile0 >> 16) & 0xFFFFu) | ((tile1 & 0xFFFFu) << 16)); // dim0 hi | dim1 lo
  g1[3] = (int)(((tile1 >> 16) & 0xFFFFu) | ((tile0 & 0xFFFFu) << 16)); // dim1 hi | tile_dim0
  g1[4] = (int)(tile1 & 0xFFFFu);                           // tile_dim1 (tile_dim2=0)
  g1[5] = (int)stride0_elems;                               // tensor_dim0_stride lo32
  g1[6] = 0;                                                // stride hi16 | dim1_stride lo16
  g1[7] = 0;
  i32x4 z4 = {0, 0, 0, 0};
  i32x8 z8 = {0, 0, 0, 0, 0, 0, 0, 0};
  __builtin_amdgcn_tensor_load_to_lds(g0, g1, z4, z4, z8, 0); // 6-arg form
}

// ---------------------------------------------------------------------------
// Kernel 1: per-batch top-16 of y_star + gather of the 16 selected points.
// ---------------------------------------------------------------------------
__global__ __launch_bounds__(256)
void hodge_topk_gather(const float* __restrict__ y, const float* __restrict__ cloud,
                       float* __restrict__ actOut, float* __restrict__ clOut) {
  int b = blockIdx.x, tid = threadIdx.x;
  const float* yr = y + (size_t)b * NPTS;
  __shared__ float sval[256];
  __shared__ int   sidx[256];
  __shared__ int   sel[16];
  __shared__ float selv[16];

  __builtin_prefetch(yr + tid, 0, 0);   // global_prefetch_b8 (CDNA5)

  for (int it = 0; it < KPTS; ++it) {
    float best = -INFINITY; int bidx = NPTS;
    for (int i = tid; i < NPTS; i += 256) {
      float v = yr[i];
      bool ex = false;
      for (int s = 0; s < it; ++s) if (sel[s] == i) { ex = true; break; }
      if (!ex && (v > best || (v == best && i < bidx))) { best = v; bidx = i; }
    }
    sval[tid] = best; sidx[tid] = bidx;
    __syncthreads();
    for (int off = 128; off > 0; off >>= 1) {
      if (tid < off) {
        float v2 = sval[tid + off]; int i2 = sidx[tid + off];
        if (v2 > sval[tid] || (v2 == sval[tid] && i2 < sidx[tid])) {
          sval[tid] = v2; sidx[tid] = i2;
        }
      }
      __syncthreads();
    }
    if (tid == 0) { sel[it] = sidx[0]; selv[it] = sval[0]; }
    __syncthreads();
  }
  if (tid < KPTS) actOut[b * KPTS + tid] = selv[tid];
  int s = tid >> 4, d = tid & 15;       // 256 threads = 16 pts x 16 dims
  clOut[((size_t)b * KPTS + s) * KPTS + d] =
      cloud[((size_t)b * NPTS + sel[s]) * KPTS + d];
}

// ---------------------------------------------------------------------------
// Kernel 2: masked 16x16 distance matrix + 4 stats features (+ zero padding).
// ---------------------------------------------------------------------------
__global__ __launch_bounds__(256)
void hodge_dist_stats(const float* __restrict__ act, const float* __restrict__ cl,
                      float* __restrict__ Dm, float* __restrict__ feats) {
  int b = blockIdx.x, tid = threadIdx.x;
  __shared__ float pts[256];
  __shared__ float sa[16];
  __shared__ float red[256];
  __shared__ float s_mean, s_max, s_tsum;
  pts[tid] = cl[(size_t)b * 256 + tid];
  if (tid < 16) sa[tid] = act[b * 16 + tid];
  __syncthreads();
  int i = tid >> 4, j = tid & 15;
  float mi = sa[i] > 0.001f ? 1.f : 0.f;
  float mj = sa[j] > 0.001f ? 1.f : 0.f;
  float sq = 0.f;
#pragma unroll
  for (int d = 0; d < 16; ++d) { float df = pts[i*16+d] - pts[j*16+d]; sq += df * df; }
  float dv = (sq > 1e-12f) ? sqrtf(sq) : 0.f;
  dv *= mi * mj;
  Dm[(size_t)b * 256 + tid] = dv;
  float tri = (j > i) ? (mi * mj) : 0.f;

  red[tid] = tri; __syncthreads();
  for (int off = 128; off > 0; off >>= 1) { if (tid < off) red[tid] += red[tid+off]; __syncthreads(); }
  if (tid == 0) s_tsum = fmaxf(red[0], 1.f);
  __syncthreads();

  red[tid] = dv * tri; __syncthreads();
  for (int off = 128; off > 0; off >>= 1) { if (tid < off) red[tid] += red[tid+off]; __syncthreads(); }
  if (tid == 0) s_mean = red[0] / s_tsum;
  __syncthreads();

  red[tid] = dv * tri; __syncthreads();
  for (int off = 128; off > 0; off >>= 1) { if (tid < off) red[tid] = fmaxf(red[tid], red[tid+off]); __syncthreads(); }
  if (tid == 0) s_max = red[0];
  __syncthreads();

  float mean = s_mean;
  float diff = dv - mean;
  red[tid] = diff * diff * tri; __syncthreads();
  for (int off = 128; off > 0; off >>= 1) { if (tid < off) red[tid] += red[tid+off]; __syncthreads(); }
  if (tid == 0) {
    float var = red[0] / s_tsum;
    float* f = feats + b * FDIM_PAD;
    f[24] = mean; f[25] = s_max; f[26] = var; f[27] = mean / (s_max + 1e-6f);
    f[28] = 0.f; f[29] = 0.f; f[30] = 0.f; f[31] = 0.f;
  }
}

// ---------------------------------------------------------------------------
// Kernel 2b: zero-padded copy of w_in (28x512 -> 32x512) so the WMMA K-loop
// in the first GEMM has no divergent guarded loads.
// ---------------------------------------------------------------------------
__global__ __launch_bounds__(256)
void pad_w_in(const float* __restrict__ w_in, float* __restrict__ wpad) {
  int idx = blockIdx.x * 256 + threadIdx.x;     // 32*512 = 16384 total
  if (idx < FDIM_PAD * HID)
    wpad[idx] = (idx < 28 * HID) ? w_in[idx] : 0.f;
}

// ---------------------------------------------------------------------------
// Kernel 3: per (batch, scale): build L0 (16x16) and L1 (120x120) in LDS,
// parallel cyclic Jacobi eigensolve, write 4 smallest eigenvalues of each.
// ~61 KB LDS per workgroup (needs CDNA5's 320 KB/WGP pool).
// ---------------------------------------------------------------------------
__global__ __launch_bounds__(256)
void hodge_eig(const float* __restrict__ Dm, const float* __restrict__ act,
               const float* __restrict__ logs, float* __restrict__ feats) {
  int bs = blockIdx.x;
  int b = bs / NSC, s = bs % NSC;
  int tid = threadIdx.x;
  __shared__ float L1[NE * NE];
  __shared__ float L0[256];
  __shared__ float W1[NE];
  __shared__ float sD[256];
  __shared__ float sa[16];
  __shared__ unsigned char sei[NE], sej[NE];
  __shared__ float rc[60], rs[60];
  __shared__ int rp[60], rq[60];

  sD[tid] = Dm[(size_t)b * 256 + tid];
  if (tid < 16) sa[tid] = act[b * 16 + tid];
  if (tid == 0) {
    int n = 0;
    for (int i = 0; i < 16; ++i)
      for (int j = i + 1; j < 16; ++j) { sei[n] = (unsigned char)i; sej[n] = (unsigned char)j; ++n; }
  }
  __syncthreads();
  float sigma = expf(logs[s]);
  float denom = 2.f * sigma * sigma + 1e-8f;
  if (tid < NE) {
    int i = sei[tid], j = sej[tid];
    float mi = sa[i] > 0.001f ? 1.f : 0.f;
    float mj = sa[j] > 0.001f ? 1.f : 0.f;
    float d = sD[i * 16 + j];
    W1[tid] = expf(-(d * d) / denom) * mi * mj;
  }
  __syncthreads();

  // ---- L0 = B1 diag(W1) B1^T + tau I  (graph Laplacian of W1) ----
  {
    int k = tid >> 4, l = tid & 15;
    float v;
    if (k == l) {
      float deg = 0.f;
      for (int e = 0; e < NE; ++e) if (sei[e] == k || sej[e] == k) deg += W1[e];
      v = deg + TAU_C;
    } else {
      int a = k < l ? k : l, c = k < l ? l : k;
      v = -W1[e2i(a, c)];
    }
    L0[tid] = v;
  }
  __syncthreads();

  // ---- Jacobi on L0 (n=16): 8 disjoint pairs/round, 15 rounds/sweep ----
  for (int sweep = 0; sweep < 8; ++sweep) {
    for (int r = 0; r < 15; ++r) {
      if (tid < 8) {
        int pa = (tid == 0) ? 0 : 1 + ((tid - 1 + r) % 15);
        int posb = 15 - tid;
        int pb = 1 + ((posb - 1 + r) % 15);
        int p = pa < pb ? pa : pb, q = pa < pb ? pb : pa;
        float app = L0[p*16+p], aqq = L0[q*16+q], apq = L0[p*16+q];
        float c, sn;
        if (fabsf(apq) > 1e-30f) {
          float th = (aqq - app) / (2.f * apq);
          float t = ((th >= 0.f) ? 1.f : -1.f) / (fabsf(th) + sqrtf(th * th + 1.f));
          c = 1.f / sqrtf(1.f + t * t); sn = t * c;
        } else { c = 1.f; sn = 0.f; }
        rp[tid] = p; rq[tid] = q; rc[tid] = c; rs[tid] = sn;
      }
      __syncthreads();
      if (tid < 128) {   // row phase: 8 pairs x 16 cols
        int pi = tid >> 4, m = tid & 15;
        int p = rp[pi], q = rq[pi]; float c = rc[pi], sn = rs[pi];
        float ap = L0[p*16+m], aq = L0[q*16+m];
        L0[p*16+m] = c*ap - sn*aq; L0[q*16+m] = sn*ap + c*aq;
      }
      __syncthreads();
      if (tid < 128) {   // col phase
        int pi = tid >> 4, m = tid & 15;
        int p = rp[pi], q = rq[pi]; float c = rc[pi], sn = rs[pi];
        float ap = L0[m*16+p], aq = L0[m*16+q];
        L0[m*16+p] = c*ap - sn*aq; L0[m*16+q] = sn*ap + c*aq;
      }
      __syncthreads();
    }
  }
  if (tid < 16) {        // rank-select 4 smallest eigenvalues (ascending)
    float d = L0[tid * 17];
    int rank = 0;
    for (int f = 0; f < 16; ++f) {
      float df = L0[f * 17];
      if (df < d || (df == d && f < tid)) ++rank;
    }
    if (rank < 4) feats[b * FDIM_PAD + s * 8 + rank] = d;
  }
  __syncthreads();

  // ---- L1 = B1^T diag(act) B1 + B2 diag(W2) B2^T + tau I ----
  for (int idx = tid; idx < NE * NE; idx += 256) {
    int e = idx / NE, f = idx % NE;
    int i = sei[e], j = sej[e], p = sei[f], q = sej[f];
    float v = 0.f;
    if (i == p) v += sa[i];
    if (i == q) v -= sa[i];
    if (j == p) v -= sa[j];
    if (j == q) v += sa[j];
    if (e == f) v += TAU_C;
    L1[idx] = v;
  }
  __syncthreads();
  // up-term diagonal: deterministic per-edge gather over its 14 triangles
  if (tid < NE) {
    int i = sei[tid], j = sej[tid];
    float we = W1[tid], acc = 0.f;
    for (int k = 0; k < 16; ++k) {
      if (k == i || k == j) continue;
      int a1 = i < k ? i : k, b1 = i < k ? k : i;
      int a2 = j < k ? j : k, b2 = j < k ? k : j;
      acc += we * W1[e2i(a1, b1)] * W1[e2i(a2, b2)];
    }
    L1[tid * NE + tid] += acc;
  }
  __syncthreads();
  // up-term off-diagonal: each (e,f) entry belongs to exactly one triangle
  {
    int t = 0;
    for (int i = 0; i < 16; ++i)
      for (int j = i + 1; j < 16; ++j)
        for (int k = j + 1; k < 16; ++k) {
          if ((t % 256) == tid) {
            int eij = e2i(i, j), ejk = e2i(j, k), eik = e2i(i, k);
            float w = W1[eij] * W1[ejk] * W1[eik];   // W2 for this triangle
            L1[eij*NE+ejk] += w; L1[ejk*NE+eij] += w;
            L1[eij*NE+eik] -= w; L1[eik*NE+eij] -= w;
            L1[ejk*NE+eik] -= w; L1[eik*NE+ejk] -= w;
          }
          ++t;
        }
  }
  __syncthreads();

  // ---- Jacobi on L1 (n=120): 60 disjoint pairs/round, 119 rounds/sweep ----
  for (int sweep = 0; sweep < 8; ++sweep) {
    for (int r = 0; r < 119; ++r) {
      if (tid < 60) {
        int pa = (tid == 0) ? 0 : 1 + ((tid - 1 + r) % 119);
        int posb = 119 - tid;
        int pb = 1 + ((posb - 1 + r) % 119);
        int p = pa < pb ? pa : pb, q = pa < pb ? pb : pa;
        float app = L1[p*NE+p], aqq = L1[q*NE+q], apq = L1[p*NE+q];
        float c, sn;
        if (fabsf(apq) > 1e-30f) {
          float th = (aqq - app) / (2.f * apq);
          float t = ((th >= 0.f) ? 1.f : -1.f) / (fabsf(th) + sqrtf(th * th + 1.f));
          c = 1.f / sqrtf(1.f + t * t); sn = t * c;
        } else { c = 1.f; sn = 0.f; }
        rp[tid] = p; rq[tid] = q; rc[tid] = c; rs[tid] = sn;
      }
      __syncthreads();
      for (int task = tid; task < 60 * NE; task += 256) {  // row phase
        int pi = task / NE, m = task % NE;
        int p = rp[pi], q = rq[pi]; float c = rc[pi], sn = rs[pi];
        float ap = L1[p*NE+m], aq = L1[q*NE+m];
        L1[p*NE+m] = c*ap - sn*aq; L1[q*NE+m] = sn*ap + c*aq;
      }
      __syncthreads();
      for (int task = tid; task < 60 * NE; task += 256) {  // col phase
        int pi = task / NE, m = task % NE;
        int p = rp[pi], q = rq[pi]; float c = rc[pi], sn = rs[pi];
        float ap = L1[m*NE+p], aq = L1[m*NE+q];
        L1[m*NE+p] = c*ap - sn*aq; L1[m*NE+q] = sn*ap + c*aq;
      }
      __syncthreads();
    }
  }
  if (tid < NE) {
    float d = L1[tid * NE + tid];
    int rank = 0;
    for (int f = 0; f < NE; ++f) {
      float df = L1[f * NE + f];
      if (df < d || (df == d && f < tid)) ++rank;
    }
    if (rank < 4) feats[b * FDIM_PAD + s * 8 + 4 + rank] = d;
  }
}

// ---------------------------------------------------------------------------
// Kernel 4: h = gelu(feats(256x32) @ w_in_pad(32x512) + b_in) via f32 WMMA.
// One wave32 per 16x16 tile; branch-free K-loop (weights pre-padded).
// A layout (16x4 f32): lane l holds M=l&15, K=(l>>4)*2 + {0,1}.
// C layout: lane l holds N=l&15, VGPR v -> M = v + 8*(l>>4).
// ---------------------------------------------------------------------------
__global__ __launch_bounds__(32)
void mlp_in_wmma(const float* __restrict__ A, const float* __restrict__ W,
                 const float* __restrict__ bias, float* __restrict__ H) {
  int lane = threadIdx.x;
  int m = lane & 15;
  int kh = (lane >> 4) << 1;
  int rowA = blockIdx.x * 16 + m;
  int colB = blockIdx.y * 16 + m;
  v8f acc = {};
  for (int k0 = 0; k0 < FDIM_PAD; k0 += 4) {
    v2f a, bv;
    a.x = A[rowA * FDIM_PAD + k0 + kh];
    a.y = A[rowA * FDIM_PAD + k0 + kh + 1];
    bv.x = W[(k0 + kh) * HID + colB];
    bv.y = W[(k0 + kh + 1) * HID + colB];
    acc = __builtin_amdgcn_wmma_f32_16x16x4_f32(false, a, false, bv,
                                                (short)0, acc, false, false);
  }
  int half = lane >> 4;
#pragma unroll
  for (int v = 0; v < 8; ++v) {
    int rM = blockIdx.x * 16 + v + half * 8;
    int cN = blockIdx.y * 16 + m;
    float x = acc[v] + bias[cN];
    float g = 0.5f * x * (1.f + erff(x * 0.7071067811865476f));  // exact GELU
    H[rM * HID + cN] = g;
  }
}

// ---------------------------------------------------------------------------
// Kernel 5: out = h(256x512) @ w_out(512x512) + b_out via f32 WMMA, with the
// A slab (16x512) and strided B slab (512x16) staged into LDS by the Tensor
// Data Mover; s_wait_tensorcnt gates the WMMA loop. 64 KB LDS / workgroup.
// ---------------------------------------------------------------------------
__global__ __launch_bounds__(32)
void mlp_out_wmma(const float* __restrict__ H, const float* __restrict__ W,
                  const float* __restrict__ bias, float* __restrict__ Out) {
  __shared__ float ldsA[16 * HID];   // 32 KB: 16 rows of h, full K
  __shared__ float ldsB[HID * 16];   // 32 KB: 512 rows x 16 cols of w_out
  int lane = threadIdx.x;
  int rowBase = blockIdx.x * 16;
  int colBase = blockIdx.y * 16;

  // TDM: A tile = 512 contiguous elems x 16 rows (stride HID);
  //      B tile = 16 elems x 512 rows (stride HID).
  tdm_load_2d_f32(&ldsA[0], H + (size_t)rowBase * HID, HID, 16, HID);
  tdm_load_2d_f32(&ldsB[0], W + colBase, 16, HID, HID);
  __builtin_amdgcn_s_wait_tensorcnt(0);

  int m = lane & 15;
  int kh = (lane >> 4) << 1;
  v8f acc = {};
  for (int k0 = 0; k0 < HID; k0 += 4) {
    v2f a, bv;
    a.x = ldsA[m * HID + k0 + kh];
    a.y = ldsA[m * HID + k0 + kh + 1];
    bv.x = ldsB[(k0 + kh) * 16 + m];
    bv.y = ldsB[(k0 + kh + 1) * 16 + m];
    acc = __builtin_amdgcn_wmma_f32_16x16x4_f32(false, a, false, bv,
                                                (short)0, acc, false, false);
  }
  int half = lane >> 4;
#pragma unroll
  for (int v = 0; v < 8; ++v) {
    int rM = rowBase + v + half * 8;
    int cN = colBase + m;
    Out[rM * HID + cN] = acc[v] + bias[cN];
  }
}

// ---------------------------------------------------------------------------
extern "C" void kernel_launch(void* const* d_in, const int* in_sizes, int n_in,
                              void* d_out, int out_size, void* d_ws, size_t ws_size,
                              hipStream_t stream) {
  const float* cloud  = (const float*)d_in[0];   // (256, 8192, 16)
  const float* ystar  = (const float*)d_in[1];   // (256, 8192)
  const float* logsc  = (const float*)d_in[2];   // (3,)
  const float* w_in   = (const float*)d_in[3];   // (28, 512)
  const float* b_in   = (const float*)d_in[4];   // (512,)
  const float* w_out  = (const float*)d_in[5];   // (512, 512)
  const float* b_out  = (const float*)d_in[6];   // (512,)
  float* out = (float*)d_out;                    // (256, 512)

  float* ws    = (float*)d_ws;
  float* act   = ws;                      // 256*16
  float* cl    = act + 256 * 16;          // 256*256
  float* Dm    = cl + 256 * 256;          // 256*256
  float* feats = Dm + 256 * 256;          // 256*32 (zero-padded K)
  float* H     = feats + 256 * FDIM_PAD;  // 256*512
  float* wpad  = H + 256 * HID;           // 32*512 zero-padded w_in

  hodge_topk_gather<<<256, 256, 0, stream>>>(ystar, cloud, act, cl);
  hodge_dist_stats <<<256, 256, 0, stream>>>(act, cl, Dm, feats);
  pad_w_in         <<<(FDIM_PAD * HID + 255) / 256, 256, 0, stream>>>(w_in, wpad);
  hodge_eig        <<<256 * NSC, 256, 0, stream>>>(Dm, act, logsc, feats);
  dim3 g1(16, HID / 16);
  mlp_in_wmma <<<g1, 32, 0, stream>>>(feats, wpad, b_in, H);
  dim3 g2(16, HID / 16);
  mlp_out_wmma<<<g2, 32, 0, stream>>>(H, w_out, b_out, out);
}